// Net_36799279793008
// MI455X (gfx1250) — compile-verified
//
#include <hip/hip_runtime.h>
#include <math.h>

typedef __attribute__((ext_vector_type(2))) float v2f;
typedef __attribute__((ext_vector_type(8))) float v8f;

#define IN_C  128
#define HID_C 128
#define OUT_C 64

// ---------------- degree / normalization ----------------

__global__ void deg_init_kernel(float* __restrict__ deg, int n) {
    int i = blockIdx.x * blockDim.x + threadIdx.x;
    if (i < n) deg[i] = 1.0f;               // self-loop contributes 1
}

__global__ void deg_accum_kernel(const long long* __restrict__ dst,
                                 float* __restrict__ deg, int E) {
    int e = blockIdx.x * blockDim.x + threadIdx.x;
    if (e < E) atomicAdd(&deg[dst[e]], 1.0f);
}

__global__ void dinv_kernel(float* __restrict__ deg, int n) {
    int i = blockIdx.x * blockDim.x + threadIdx.x;
    if (i < n) {
        float d = deg[i];
        deg[i] = (d > 0.0f) ? rsqrtf(d) : 0.0f;   // in place: deg -> deg^-1/2
    }
}

// ---------------- fp32 WMMA GEMM: C[M,N] = A[M,K] @ B[K,N] ----------------
// One wave computes one 16x16 tile of C, accumulating K in steps of 4 via
// V_WMMA_F32_16X16X4_F32.  M need not be a multiple of 16 only if padded;
// here M = 100000 = 6250*16 exactly.  N,K multiples of 16.

__global__ void gemm_wmma_f32_kernel(const float* __restrict__ A,
                                     const float* __restrict__ B,
                                     float* __restrict__ C,
                                     int M, int N, int K) {
    const int wave = threadIdx.x >> 5;
    const int lane = threadIdx.x & 31;
    const int tiles_n = N >> 4;
    const int t  = blockIdx.x * (blockDim.x >> 5) + wave;
    const int tm = t / tiles_n;
    const int tn = t - tm * tiles_n;
    if (tm * 16 >= M) return;               // wave-uniform: EXEC stays all-1s

    const int half = lane >> 4;             // 0: K pair {0,1}, 1: K pair {2,3}
    const int lm   = lane & 15;             // A row / B col / C col within tile

    const float* Arow  = A + (size_t)(tm * 16 + lm) * K + 2 * half;
    const float* Bbase = B + (size_t)(2 * half) * N + tn * 16 + lm;

    v8f acc = {};
    #pragma unroll
    for (int kk = 0; kk < 128; kk += 4) {   // K is 128 in both layers
        if (kk >= K) break;
        v2f a = *(const v2f*)(Arow + kk);                   // A[m][kk+2h : kk+2h+1]
        v2f b;
        b.x = Bbase[(size_t)kk * N];                        // B[kk+2h  ][n]
        b.y = Bbase[(size_t)kk * N + N];                    // B[kk+2h+1][n]
        acc = __builtin_amdgcn_wmma_f32_16x16x4_f32(
            /*neg_a=*/false, a, /*neg_b=*/false, b,
            /*c_mod=*/(short)0, acc, /*reuse_a=*/false, /*reuse_b=*/false);
    }

    // C/D layout: VGPR r -> row (r + 8*half), col = lm
    float* Crow = C + (size_t)(tm * 16 + 8 * half) * N + tn * 16 + lm;
    #pragma unroll
    for (int r = 0; r < 8; ++r) Crow[(size_t)r * N] = acc[r];
}

// ---------------- aggregation ----------------

// out[n][c] = dinv[n]^2 * h[n][c]   (self-loop term; initializes the buffer)
__global__ void selfloop_init_kernel(const float* __restrict__ h,
                                     const float* __restrict__ dinv,
                                     float* __restrict__ out,
                                     long long total, int C) {
    long long i = (long long)blockIdx.x * blockDim.x + threadIdx.x;
    if (i >= total) return;
    int n = (int)(i / C);
    float di = dinv[n];
    out[i] = h[i] * di * di;
}

// atomicAdd(out[dst][c], h[src][c] * dinv[src]*dinv[dst]); 4 channels/thread,
// 32 (C=128) or 16 (C=64) consecutive lanes cover one edge -> coalesced.
__global__ void agg_edges_kernel(const long long* __restrict__ src,
                                 const long long* __restrict__ dst,
                                 const float* __restrict__ dinv,
                                 const float* __restrict__ h,
                                 float* __restrict__ out,
                                 int E, int gshift, int C) {
    long long tid = (long long)blockIdx.x * blockDim.x + threadIdx.x;
    long long total = (long long)E << gshift;
    if (tid >= total) return;
    int e = (int)(tid >> gshift);
    int g = (int)(tid & ((1 << gshift) - 1));
    long long s = src[e];
    long long d = dst[e];
    float w = dinv[s] * dinv[d];
    const float4 hv = *(const float4*)(h + (size_t)s * C + 4 * g);
    float* o = out + (size_t)d * C + 4 * g;
    atomicAdd(o + 0, hv.x * w);
    atomicAdd(o + 1, hv.y * w);
    atomicAdd(o + 2, hv.z * w);
    atomicAdd(o + 3, hv.w * w);
}

// out[i] = (relu?) (out[i] + b[i % C])
__global__ void bias_act_kernel(float* __restrict__ out,
                                const float* __restrict__ b,
                                long long total, int C, int do_relu) {
    long long i = (long long)blockIdx.x * blockDim.x + threadIdx.x;
    if (i >= total) return;
    int c = (int)(i % C);
    float v = out[i] + b[c];
    out[i] = do_relu ? fmaxf(v, 0.0f) : v;
}

// ---------------- decoder: dot(z[a], z[b]) over 64 channels ----------------

__global__ void decode_kernel(const long long* __restrict__ ea,
                              const long long* __restrict__ eb,
                              const float* __restrict__ z,
                              float* __restrict__ out, int L) {
    int wave = threadIdx.x >> 5;
    int lane = threadIdx.x & 31;
    int p = blockIdx.x * (blockDim.x >> 5) + wave;
    if (p >= L) return;
    long long a = ea[p], b = eb[p];
    const float* za = z + (size_t)a * OUT_C;
    const float* zb = z + (size_t)b * OUT_C;
    float partial = za[lane] * zb[lane] + za[lane + 32] * zb[lane + 32];
    #pragma unroll
    for (int off = 16; off > 0; off >>= 1)
        partial += __shfl_xor(partial, off, 32);
    if (lane == 0) out[p] = partial;
}

// ---------------- launcher ----------------

static inline unsigned cdivu(long long a, long long b) {
    return (unsigned)((a + b - 1) / b);
}

extern "C" void kernel_launch(void* const* d_in, const int* in_sizes, int n_in,
                              void* d_out, int out_size, void* d_ws, size_t ws_size,
                              hipStream_t stream) {
    const float*     x   = (const float*)d_in[0];
    const long long* ei  = (const long long*)d_in[1];
    const long long* eli = (const long long*)d_in[2];
    const float*     W1  = (const float*)d_in[3];
    const float*     b1  = (const float*)d_in[4];
    const float*     W2  = (const float*)d_in[5];
    const float*     b2  = (const float*)d_in[6];
    float*           out = (float*)d_out;

    const int N = in_sizes[0] / IN_C;     // 100000 nodes
    const int E = in_sizes[1] / 2;        // 1.6M edges
    const int L = out_size;               // 200000 label edges

    const long long* src = ei;
    const long long* dst = ei + E;
    const long long* la  = eli;
    const long long* lb  = eli + L;

    char*  ws   = (char*)d_ws;
    float* bufA = (float*)ws;                                    // N*HID_C f32
    float* bufB = (float*)(ws + (size_t)N * HID_C * sizeof(float));
    float* dinv = (float*)(ws + 2 * (size_t)N * HID_C * sizeof(float)); // N f32

    const int TB = 256;

    // degree -> dinv (in place)
    deg_init_kernel<<<cdivu(N, TB), TB, 0, stream>>>(dinv, N);
    deg_accum_kernel<<<cdivu(E, TB), TB, 0, stream>>>(dst, dinv, E);
    dinv_kernel<<<cdivu(N, TB), TB, 0, stream>>>(dinv, N);

    // ---- layer 1 ----
    // bufA = x @ W1
    {
        long long tiles = (long long)cdivu(N, 16) * (HID_C / 16);
        gemm_wmma_f32_kernel<<<cdivu(tiles, 8), TB, 0, stream>>>(x, W1, bufA, N, HID_C, IN_C);
    }
    long long tot1 = (long long)N * HID_C;
    selfloop_init_kernel<<<cdivu(tot1, TB), TB, 0, stream>>>(bufA, dinv, bufB, tot1, HID_C);
    agg_edges_kernel<<<cdivu((long long)E * (HID_C / 4), TB), TB, 0, stream>>>(
        src, dst, dinv, bufA, bufB, E, /*gshift=*/5, HID_C);
    bias_act_kernel<<<cdivu(tot1, TB), TB, 0, stream>>>(bufB, b1, tot1, HID_C, /*relu=*/1);

    // ---- layer 2 ----
    // bufA (reused) = bufB @ W2
    {
        long long tiles = (long long)cdivu(N, 16) * (OUT_C / 16);
        gemm_wmma_f32_kernel<<<cdivu(tiles, 8), TB, 0, stream>>>(bufB, W2, bufA, N, OUT_C, HID_C);
    }
    long long tot2 = (long long)N * OUT_C;
    selfloop_init_kernel<<<cdivu(tot2, TB), TB, 0, stream>>>(bufA, dinv, bufB, tot2, OUT_C);
    agg_edges_kernel<<<cdivu((long long)E * (OUT_C / 4), TB), TB, 0, stream>>>(
        src, dst, dinv, bufA, bufB, E, /*gshift=*/4, OUT_C);
    bias_act_kernel<<<cdivu(tot2, TB), TB, 0, stream>>>(bufB, b2, tot2, OUT_C, /*relu=*/0);

    // ---- decoder ----
    decode_kernel<<<cdivu(L, TB / 32), TB, 0, stream>>>(la, lb, bufB, out, L);
}